// VPForecaster_30477087933036
// MI455X (gfx1250) — compile-verified
//
#include <hip/hip_runtime.h>
#include <hip/hip_bf16.h>

typedef __attribute__((ext_vector_type(2))) float v2f;
typedef __attribute__((ext_vector_type(8))) float v8f;

#define B_DIM 4096
#define T_DIM 256
#define C_DIM 64
#define H_DIM 16
#define S_DIM 10

__device__ __forceinline__ v8f wmma4(v2f a, v2f b, v8f c) {
    return __builtin_amdgcn_wmma_f32_16x16x4_f32(false, a, false, b, (short)0, c, false, false);
}

// Hardware tanh (TRANS op on CDNA5) with safe fallback.
#if __has_builtin(__builtin_amdgcn_tanhf)
__device__ __forceinline__ float ftanh(float x) { return __builtin_amdgcn_tanhf(x); }
#elif __has_builtin(__builtin_amdgcn_tanh_f32)
__device__ __forceinline__ float ftanh(float x) { return __builtin_amdgcn_tanh_f32(x); }
#else
__device__ __forceinline__ float ftanh(float x) {
    float e = __expf(-2.0f * fabsf(x));
    float r = (1.0f - e) / (1.0f + e);
    return copysignf(r, x);
}
#endif

__device__ __forceinline__ float fsig(float x) {
    return 1.0f / (1.0f + __expf(-x));
}

// Build B-fragments of M^T (K=16) from M^T held in C/D layout (hD) plus its
// lane-xor-16 swapped copy (hS).  Derivation: h^T[k][n] lives in vgpr k%8 on
// lane (k/8)*16+n; chunk kk needs rows k0=4kk+2*half, k0+1.
__device__ __forceinline__ void bfrag_from_D(const float hD[8], const float hS[8],
                                             int half, v2f b[4]) {
    b[0].x = half ? hS[2] : hD[0];  b[0].y = half ? hS[3] : hD[1];
    b[1].x = half ? hS[6] : hD[4];  b[1].y = half ? hS[7] : hD[5];
    b[2].x = half ? hD[2] : hS[0];  b[2].y = half ? hD[3] : hS[1];
    b[3].x = half ? hD[6] : hS[4];  b[3].y = half ? hD[7] : hS[5];
}

// ---------------------------------------------------------------------------
// Kernel 1: fused LayerNorm(C=64) + GEMM (B*T,64)x(64,16) + bias.
// One wave per 16-row tile; output transposed to (T, B, 16).
// ---------------------------------------------------------------------------
__global__ __launch_bounds__(256)
void ln_xw_kernel(const float* __restrict__ cov, const float* __restrict__ gamma,
                  const float* __restrict__ beta, const float* __restrict__ wx,
                  const float* __restrict__ rb, float* __restrict__ xw) {
    const int lane = threadIdx.x & 31;
    const int wave = threadIdx.x >> 5;
    const int tile = blockIdx.x * 8 + wave;          // 65536 tiles
    const int n    = lane & 15;
    const int half = lane >> 4;

    const long rowM = (long)tile * 16 + n;
    const float* crow = cov + rowM * C_DIM;

    v2f xa[16];
    float sum = 0.0f, sumsq = 0.0f;
#pragma unroll
    for (int kk = 0; kk < 16; ++kk) {
        v2f v = *(const v2f*)(crow + 4 * kk + 2 * half);
        xa[kk] = v;
        sum   += v.x + v.y;
        sumsq += v.x * v.x + v.y * v.y;
    }
    sum   += __shfl_xor(sum, 16, 32);
    sumsq += __shfl_xor(sumsq, 16, 32);
    const float mean = sum * (1.0f / 64.0f);
    const float var  = sumsq * (1.0f / 64.0f) - mean * mean;
    const float rs   = rsqrtf(var + 1e-3f);

    v8f c;
    const float bias = rb[n];
#pragma unroll
    for (int r = 0; r < 8; ++r) c[r] = bias;

#pragma unroll
    for (int kk = 0; kk < 16; ++kk) {
        const int k0 = 4 * kk + 2 * half;
        v2f gv = *(const v2f*)(gamma + k0);
        v2f bv = *(const v2f*)(beta + k0);
        v2f a;
        a.x = (xa[kk].x - mean) * rs * gv.x + bv.x;
        a.y = (xa[kk].y - mean) * rs * gv.y + bv.y;
        v2f b;
        b.x = wx[(k0 + 0) * H_DIM + n];
        b.y = wx[(k0 + 1) * H_DIM + n];
        c = wmma4(a, b, c);
    }

    const int baseM = tile * 16 + 8 * half;
#pragma unroll
    for (int r = 0; r < 8; ++r) {
        const int M2 = baseM + r;          // b = M2/256, t = M2%256
        const int bb = M2 >> 8;
        const int tt = M2 & 255;
        xw[((long)tt * B_DIM + bb) * H_DIM + n] = c[r];
    }
}

// ---------------------------------------------------------------------------
// Kernel 2: RNN scan only, transposed: h^T = tanh(xw^T + Wh^T · h_prev^T).
// A-fragments (Wh^T) constant; h relayout D->B via lane-xor-16 shuffles.
// One wave per 16-batch tile, no LDS.
// ---------------------------------------------------------------------------
__global__ __launch_bounds__(32)
void rnn_kernel(const float* __restrict__ xw, const float* __restrict__ wh,
                float* __restrict__ hs) {
    const int lane = threadIdx.x & 31;
    const int n    = lane & 15;
    const int half = lane >> 4;
    const int b0   = blockIdx.x * 16;                // 256 blocks

    v2f aWh[4];
#pragma unroll
    for (int kk = 0; kk < 4; ++kk) {
        const int k0 = 4 * kk + 2 * half;
        aWh[kk].x = wh[(k0 + 0) * H_DIM + n];        // Wh^T[M=n][K=k0..]
        aWh[kk].y = wh[(k0 + 1) * H_DIM + n];
    }

    float hD[8];
#pragma unroll
    for (int r = 0; r < 8; ++r) hD[r] = 0.0f;

    for (int t = 0; t < T_DIM; ++t) {
        // accumulator = xw^T tile: lane reads 32 contiguous bytes
        v8f c;
        const float* xp = xw + ((long)t * B_DIM + b0 + n) * H_DIM + 8 * half;
#pragma unroll
        for (int r = 0; r < 8; ++r) c[r] = xp[r];

        float hS[8];
#pragma unroll
        for (int r = 0; r < 8; ++r) hS[r] = __shfl_xor(hD[r], 16, 32);
        v2f bf[4];
        bfrag_from_D(hD, hS, half, bf);
#pragma unroll
        for (int kk = 0; kk < 4; ++kk) c = wmma4(aWh[kk], bf[kk], c);

#pragma unroll
        for (int r = 0; r < 8; ++r) hD[r] = ftanh(c[r]);

        float* hp = hs + ((long)t * B_DIM + b0 + n) * H_DIM + 8 * half;
#pragma unroll
        for (int r = 0; r < 8; ++r) hp[r] = hD[r];
    }
}

// ---------------------------------------------------------------------------
// Kernel 3: drivers = tanh(hs @ d1_w + b1) @ d2_w + b2, fully parallel over
// all (t,b) rows, transposed form (weights as constant A-fragments; GEMM1->
// GEMM2 chained via shuffle relayout, no LDS). Writes (T, B, 10).
// ---------------------------------------------------------------------------
__global__ __launch_bounds__(256)
void drivers_kernel(const float* __restrict__ hs, const float* __restrict__ d1w,
                    const float* __restrict__ d1b, const float* __restrict__ d2w,
                    const float* __restrict__ d2b, float* __restrict__ drv) {
    const int lane = threadIdx.x & 31;
    const int wave = threadIdx.x >> 5;
    const int tile = blockIdx.x * 8 + wave;          // 65536 tiles
    const int n    = lane & 15;
    const int half = lane >> 4;
    const long flat0 = (long)tile * 16;

    v2f a1[4], a2[4];
#pragma unroll
    for (int kk = 0; kk < 4; ++kk) {
        const int k0 = 4 * kk + 2 * half;
        a1[kk].x = d1w[(k0 + 0) * H_DIM + n];  a1[kk].y = d1w[(k0 + 1) * H_DIM + n];
        a2[kk].x = d2w[(k0 + 0) * H_DIM + n];  a2[kk].y = d2w[(k0 + 1) * H_DIM + n];
    }

    // B-fragments of hs^T for this tile: per-lane b64 loads
    v2f bh[4];
#pragma unroll
    for (int kk = 0; kk < 4; ++kk)
        bh[kk] = *(const v2f*)(hs + (flat0 + n) * H_DIM + 4 * kk + 2 * half);

    // u^T = tanh(d1w^T · hs^T + b1)
    v8f c;
    const float* p1 = d1b + 8 * half;
#pragma unroll
    for (int r = 0; r < 8; ++r) c[r] = p1[r];
#pragma unroll
    for (int kk = 0; kk < 4; ++kk) c = wmma4(a1[kk], bh[kk], c);

    float uD[8], uS[8];
#pragma unroll
    for (int r = 0; r < 8; ++r) uD[r] = ftanh(c[r]);
#pragma unroll
    for (int r = 0; r < 8; ++r) uS[r] = __shfl_xor(uD[r], 16, 32);
    v2f bu[4];
    bfrag_from_D(uD, uS, half, bu);

    // drv^T = d2w^T · u^T + b2
    v8f c2;
    const float* p2 = d2b + 8 * half;
#pragma unroll
    for (int r = 0; r < 8; ++r) c2[r] = p2[r];
#pragma unroll
    for (int kk = 0; kk < 4; ++kk) c2 = wmma4(a2[kk], bu[kk], c2);

    // store first 10 driver components: drv[(t*B+b)*10 + comp]
    const long row = flat0 + n;
#pragma unroll
    for (int r = 0; r < 8; ++r) {
        const int comp = r + 8 * half;
        if (comp < 10) drv[row * 10 + comp] = c2[r];
    }
}

// ---------------------------------------------------------------------------
// Kernel 4: cash-budget scan. One lane per batch row; drivers laid out
// (T, B, 10) so each step's loads are wave-coalesced and state-independent.
// ---------------------------------------------------------------------------
__global__ __launch_bounds__(32)
void cash_kernel(const float* __restrict__ drv, const float* __restrict__ states,
                 float* __restrict__ st_out, float* __restrict__ is_out) {
    const int b = blockIdx.x * 32 + threadIdx.x;     // 128 blocks -> 4096 rows

    float s[10];
#pragma unroll
    for (int i = 0; i < 10; ++i)
        s[i] = states[(long)b * (T_DIM * S_DIM) + i];    // states_seq[b,0,:]

    float* so = st_out + (long)b * (T_DIM - 1) * S_DIM;
    float* io = is_out + (long)b * (T_DIM - 1) * 7;

    for (int t = 1; t < T_DIM; ++t) {
        const float* d = drv + ((long)t * B_DIM + b) * 10;
        const float g      = ftanh(d[0]) * 0.2f;
        const float cogs_m = fsig(d[1]);
        const float opex_m = fsig(d[2]) * 0.5f;
        const float dep_r  = fsig(d[3]) * 0.1f;
        const float capx_r = fsig(d[4]) * 0.2f;
        const float dso    = fsig(d[5]);
        const float dio    = fsig(d[6]);
        const float dpo    = fsig(d[7]);
        const float tax_r  = fsig(d[8]) * 0.35f;
        const float int_r  = fsig(d[9]) * 0.1f;

        const float rev      = s[0] * (1.0f + g);
        const float cogs     = cogs_m * rev;
        const float opex     = opex_m * rev;
        const float dep      = dep_r * s[4];
        const float interest = int_r * s[6];
        const float ebt      = rev - cogs - opex - dep - interest;
        const float tax      = tax_r * fmaxf(ebt, 0.0f);
        const float ni       = ebt - tax;
        const float ar_new   = dso * rev;
        const float inv_new  = dio * cogs;
        const float ap_new   = dpo * cogs;
        const float capex    = capx_r * rev;
        const float ppe_new  = s[4] + capex - dep;
        const float cash_new = s[1] + ni + dep - capex - (ar_new - s[2])
                               - (inv_new - s[3]) + (ap_new - s[5]);
        const float re_new   = s[8] + ni;

        s[0] = rev;  s[1] = cash_new; s[2] = ar_new; s[3] = inv_new;
        s[4] = ppe_new; s[5] = ap_new; s[8] = re_new;

        so[0] = rev;     so[1] = cash_new; so[2] = ar_new; so[3] = inv_new;
        so[4] = ppe_new; so[5] = ap_new;   so[6] = s[6];   so[7] = s[7];
        so[8] = re_new;  so[9] = s[9];
        so += S_DIM;

        io[0] = rev; io[1] = cogs; io[2] = opex; io[3] = dep;
        io[4] = interest; io[5] = tax; io[6] = ni;
        io += 7;
    }
}

extern "C" void kernel_launch(void* const* d_in, const int* in_sizes, int n_in,
                              void* d_out, int out_size, void* d_ws, size_t ws_size,
                              hipStream_t stream) {
    const float* states = (const float*)d_in[0];
    const float* cov    = (const float*)d_in[1];
    const float* gamma  = (const float*)d_in[2];
    const float* beta   = (const float*)d_in[3];
    const float* wx     = (const float*)d_in[4];
    const float* wh     = (const float*)d_in[5];
    const float* rb     = (const float*)d_in[6];
    const float* d1w    = (const float*)d_in[7];
    const float* d1b    = (const float*)d_in[8];
    const float* d2w    = (const float*)d_in[9];
    const float* d2b    = (const float*)d_in[10];

    const long nTB = (long)T_DIM * B_DIM;            // 1,048,576 rows
    float* xw_ws  = (float*)d_ws;                    // (T,B,16): 67 MB
    float* hs_ws  = xw_ws + nTB * H_DIM;             // (T,B,16): 67 MB
    float* drv_ws = xw_ws;                           // reuse xw region: (T,B,10)

    float* st_out = (float*)d_out;                               // (B,255,10)
    float* is_out = st_out + (long)B_DIM * (T_DIM - 1) * S_DIM;  // (B,255,7)

    // 65536 tiles, 8 waves/block
    ln_xw_kernel<<<dim3(8192), dim3(256), 0, stream>>>(cov, gamma, beta, wx, rb, xw_ws);

    // 256 independent recurrence chains, one wave each
    rnn_kernel<<<dim3(256), dim3(32), 0, stream>>>(xw_ws, wh, hs_ws);

    // fully parallel driver head (xw region is dead now; reuse for drivers)
    drivers_kernel<<<dim3(8192), dim3(256), 0, stream>>>(hs_ws, d1w, d1b, d2w, d2b, drv_ws);

    // 4096 scalar scan lanes
    cash_kernel<<<dim3(128), dim3(32), 0, stream>>>(drv_ws, states, st_out, is_out);
}